// AMS_52561809768904
// MI455X (gfx1250) — compile-verified
//
#include <hip/hip_runtime.h>
#include <math.h>

// ---------------- problem constants ----------------
constexpr int kNB = 32;    // batch
constexpr int kL  = 256;   // seq len
constexpr int kC  = 32;    // channels
constexpr int kD  = 32;    // hidden
constexpr int kP  = 96;    // pred len
constexpr int kT  = 528;   // multi-scale length
constexpr int kTV = 32;
constexpr int kNV = 16;
constexpr int kE  = 4;
constexpr int kFB = 129;   // rfft bins
constexpr int kCBUF = 1320; // per-expert const pack (7*32 + 1024 + 32 + 32 + 1, padded)

// GEMM staging: K is processed in chunks staged to LDS (double buffered)
constexpr int kKCHUNK = 176;               // 528 = 3 * 176, multiple of 4
constexpr int kNCHUNK = kT / kKCHUNK;      // 3
constexpr int kQUADS_PER_CHUNK = kKCHUNK * kC / 4;      // 1408 float4s
constexpr int kQUADS_PER_THREAD = kQUADS_PER_CHUNK / 128; // 11 (block = 128 threads)

typedef __attribute__((ext_vector_type(2))) float v2f;
typedef __attribute__((ext_vector_type(8))) float v8f;

// async-to-LDS builtin operand types: pointer to int4 (vector_size form, per clang's
// builtin prototype), with the LDS side carried through an AS(3)-qualified typedef.
typedef int v4i __attribute__((vector_size(16)));
typedef __attribute__((address_space(3))) v4i as3_v4i;

#if defined(__gfx1250__) && __has_builtin(__builtin_amdgcn_wmma_f32_16x16x4_f32)
#define HAS_WMMA_F32K4 1
#else
#define HAS_WMMA_F32K4 0
#endif

#if defined(__gfx1250__) && __has_builtin(__builtin_amdgcn_global_load_async_to_lds_b128) && \
    __has_builtin(__builtin_amdgcn_s_wait_asynccnt)
#define HAS_ASYNC_LDS 1
#else
#define HAS_ASYNC_LDS 0
#endif

__device__ __forceinline__ float gelu_tanh(float v) {
  const float k0 = 0.7978845608028654f; // sqrt(2/pi)
  float inner = k0 * (v + 0.044715f * v * v * v);
  return 0.5f * v * (1.0f + tanhf(inner));
}

// ---------------- stage 0: DFT amplitude spectrum ----------------
// grid (B, FBINS), block 32 (lane = channel). amp[b,f] = mean_c |rfft(x[b,:,c])[f]|
__global__ void dft_amp_kernel(const float* __restrict__ x, float* __restrict__ amp) {
  int b = blockIdx.x, f = blockIdx.y, c = threadIdx.x;
  if (f == 0) { if (c == 0) amp[b * kFB] = 0.0f; return; }  // DC suppressed
  const float* xb = x + (size_t)b * kL * kC + c;
  float re = 0.f, im = 0.f;
  float w = -6.283185307179586f * (float)f / (float)kL;
  for (int l = 0; l < kL; ++l) {
    float sv, cv;
    __sincosf(w * (float)l, &sv, &cv);
    float xv = xb[(size_t)l * kC];
    re += xv * cv;
    im += xv * sv;
  }
  float mag = sqrtf(re * re + im * im);
  for (int o = 16; o > 0; o >>= 1) mag += __shfl_down(mag, o, 32);
  if (c == 0) amp[b * kFB + f] = mag * (1.0f / (float)kC);
}

// ---------------- stage 1: noisy top-2 gating ----------------
// 1 block, 32 threads (one per batch element)
__global__ void gating_kernel(const float* __restrict__ amp, const float* __restrict__ noise,
                              const float* __restrict__ w_gate, const float* __restrict__ w_noise,
                              float* __restrict__ gates) {
  int b = threadIdx.x;
  if (b >= kNB) return;
  float logit[kE];
  for (int e = 0; e < kE; ++e) {
    float cl = 0.f, nz = 0.f;
    for (int f = 1; f < kFB; ++f) {
      float a = amp[b * kFB + f];
      cl += a * w_gate[f * kE + e];
      nz += a * w_noise[f * kE + e];
    }
    float sp = (nz > 20.f) ? nz : log1pf(expf(nz));  // softplus
    logit[e] = cl + noise[b * kE + e] * (sp + 1e-2f);
  }
  int i0 = 0;
  for (int e = 1; e < kE; ++e) if (logit[e] > logit[i0]) i0 = e;
  int i1 = -1;
  for (int e = 0; e < kE; ++e) {
    if (e == i0) continue;
    if (i1 < 0 || logit[e] > logit[i1]) i1 = e;
  }
  float g1 = 1.f / (1.f + expf(logit[i0] - logit[i1]));  // softmax over top-2
  float g0 = 1.f - g1;
  float g[kE] = {0.f, 0.f, 0.f, 0.f};
  g[i0] = g0; g[i1] = g1;
  for (int e = 0; e < kE; ++e) gates[b * kE + e] = g[e];
}

// ---------------- stage 2: A_t = softmax_axis0(relu(tv1 @ tv2)), stored TRANSPOSED ----------------
// grid (E, T) -> block handles one column w; block 256. AT[e][w][t]
__global__ void build_At_kernel(const float* __restrict__ tv1, const float* __restrict__ tv2,
                                float* __restrict__ AT) {
  int e = blockIdx.x, w = blockIdx.y, tid = threadIdx.x;
  __shared__ float col2[kTV];
  __shared__ float row[kT];
  __shared__ float red[256];
  const float* t1 = tv1 + (size_t)e * kT * kTV;
  const float* t2 = tv2 + (size_t)e * kTV * kT;
  if (tid < kTV) col2[tid] = t2[(size_t)tid * kT + w];
  __syncthreads();
  for (int t = tid; t < kT; t += 256) {
    const float* r1 = t1 + (size_t)t * kTV;
    float s = 0.f;
    #pragma unroll
    for (int v = 0; v < kTV; ++v) s += r1[v] * col2[v];
    row[t] = s > 0.f ? s : 0.f;  // relu
  }
  __syncthreads();
  float m = -1e30f;
  for (int t = tid; t < kT; t += 256) m = fmaxf(m, row[t]);
  red[tid] = m; __syncthreads();
  for (int s2 = 128; s2 > 0; s2 >>= 1) {
    if (tid < s2) red[tid] = fmaxf(red[tid], red[tid + s2]);
    __syncthreads();
  }
  m = red[0];
  __syncthreads();
  float lsum = 0.f;
  for (int t = tid; t < kT; t += 256) {
    float ev = expf(row[t] - m);
    row[t] = ev;
    lsum += ev;
  }
  red[tid] = lsum; __syncthreads();
  for (int s2 = 128; s2 > 0; s2 >>= 1) {
    if (tid < s2) red[tid] += red[tid + s2];
    __syncthreads();
  }
  float inv = 1.0f / red[0];
  float* out = AT + ((size_t)e * kT + w) * kT;
  for (int t = tid; t < kT; t += 256) out[t] = row[t] * inv;
}

// ---------------- stage 3: A_n (row softmax) + its column sums ncs1/ncs2 ----------------
// grid E, block 32 (thread = row c)
__global__ void build_An_kernel(const float* __restrict__ nv1, const float* __restrict__ nv2,
                                float* __restrict__ An, float* __restrict__ ncs1,
                                float* __restrict__ ncs2) {
  int e = blockIdx.x, c = threadIdx.x;
  __shared__ float a[kC][kC];
  __shared__ float s1[kC];
  const float* n1 = nv1 + (size_t)e * kC * kNV + (size_t)c * kNV;
  const float* n2 = nv2 + (size_t)e * kNV * kC;
  float r[kC];
  float m = -1e30f;
  for (int w = 0; w < kC; ++w) {
    float s = 0.f;
    #pragma unroll
    for (int v = 0; v < kNV; ++v) s += n1[v] * n2[v * kC + w];
    s = fmaxf(s, 0.f);
    r[w] = s;
    m = fmaxf(m, s);
  }
  float sum = 0.f;
  for (int w = 0; w < kC; ++w) { r[w] = expf(r[w] - m); sum += r[w]; }
  float inv = 1.f / sum;
  for (int w = 0; w < kC; ++w) {
    float av = r[w] * inv;
    a[c][w] = av;
    An[((size_t)e * kC + c) * kC + w] = av;
  }
  __syncthreads();
  float t1 = 0.f;
  for (int k = 0; k < kC; ++k) t1 += a[k][c];
  s1[c] = t1;
  ncs1[e * kC + c] = t1;
  __syncthreads();
  float t2 = 0.f;
  for (int k = 0; k < kC; ++k) t2 += s1[k] * a[k][c];
  ncs2[e * kC + c] = t2;
}

// ---------------- stage 4: per-expert constant folding (rank-1 MLP collapse) ----------------
// grid E, block 32 (thread = d')
__global__ void precompute_kernel(const float* __restrict__ start_w, const float* __restrict__ start_b,
                                  const float* __restrict__ tmw, const float* __restrict__ tmb,
                                  const float* __restrict__ nmw, const float* __restrict__ nmb,
                                  const float* __restrict__ out_w, const float* __restrict__ out_b,
                                  const float* __restrict__ ncs1, const float* __restrict__ ncs2,
                                  float* __restrict__ cbuf) {
  int e = blockIdx.x, d = threadIdx.x;
  const float* sw = start_w + e * kD;
  const float* sb = start_b + e * kD;
  const float* tw = tmw + (size_t)e * 3 * kD * kD;
  const float* nw = nmw + (size_t)e * 3 * kD * kD;
  float* cb = cbuf + (size_t)e * kCBUF;
  float u0 = 0, u1 = 0, u2 = 0, vs = 0, p0 = 0, p1 = 0, p2 = 0, q0 = 0, q1 = 0, q2 = 0;
  for (int j = 0; j < kD; ++j) {
    float w0t = tw[j * kD + d], w1t = tw[(kD + j) * kD + d], w2t = tw[(2 * kD + j) * kD + d];
    u0 += sw[j] * w0t; u1 += sw[j] * w1t; u2 += sw[j] * w2t;
    vs += sb[j] * (w0t + w1t + w2t);
    float w0n = nw[j * kD + d], w1n = nw[(kD + j) * kD + d], w2n = nw[(2 * kD + j) * kD + d];
    p0 += sw[j] * w0n; p1 += sw[j] * w1n; p2 += sw[j] * w2n;
    q0 += sb[j] * w0n; q1 += sb[j] * w1n; q2 += sb[j] * w2n;
  }
  cb[d] = u0; cb[32 + d] = u1; cb[64 + d] = u2; cb[96 + d] = vs + tmb[e * kD + d];
  cb[128 + d] = p0; cb[160 + d] = p1; cb[192 + d] = p2;
  float nb = nmb[e * kD + d];
  for (int c2 = 0; c2 < kC; ++c2)
    cb[224 + c2 * kD + d] = q0 + ncs1[e * kC + c2] * q1 + ncs2[e * kC + c2] * q2 + nb;
  cb[1248 + d] = out_w[e * 2 * kD + d];        // ow (time half)
  cb[1280 + d] = out_w[e * 2 * kD + kD + d];   // ow (node half)
  if (d == 0) cb[1312] = out_b[e];
}

// ---------------- stage 5: multi-scale sequence ms[e][b][t][c] ----------------
// grid (66, B, E), block 256 (T*C = 16896 = 66*256)
__global__ void build_ms_kernel(const float* __restrict__ x, float* __restrict__ ms) {
  int e = blockIdx.z, b = blockIdx.y;
  int idx = blockIdx.x * 256 + threadIdx.x;
  int t = idx >> 5, c = idx & 31;
  int s = 2 << e;  // scales 2,4,8,16
  const float* xb = x + (size_t)b * kL * kC;
  float v;
  if (t < kL) {
    v = xb[(size_t)t * kC + c];
  } else {
    int i = t - kL;
    int n = kL / s;
    if (i < n) {
      float sum = 0.f;
      for (int j = 0; j < s; ++j) sum += xb[(size_t)(i * s + j) * kC + c];
      v = sum / (float)s;
    } else {
      v = 0.f;  // pad to T
    }
  }
  ms[(((size_t)e * kNB + b) * kT + t) * kC + c] = v;
}

// ---------------- stage 6: batched GEMM  D[w,c] = sum_t AT[w,t] * B[t,c]  via f32 WMMA ----------------
// grid (B, mTileGroups, E), block 128 (4 waves); each wave -> one 16x32 output tile.
// B operand (ms[e][b], shared by all 4 waves) is staged chunk-by-chunk in LDS with
// gfx1250 async-to-LDS copies, double buffered and overlapped with WMMA compute.
__global__ void __launch_bounds__(128)
ms_matmul_wmma(const float* __restrict__ ATbase, const float* __restrict__ Bbase,
               float* __restrict__ Dbase, int mTileStart, int mTiles, int outRows) {
  __shared__ __align__(16) float sbuf[2][kKCHUNK * kC];
  int e = blockIdx.z;
  int b = blockIdx.x;
  int tid = threadIdx.x;
  int mt = blockIdx.y * (blockDim.x >> 5) + (tid >> 5);
  bool valid = (mt < mTiles);  // invalid waves still co-load + hit barriers
  int lane = tid & 31;
  int half = lane >> 4;  // 0: lanes 0-15, 1: lanes 16-31
  int l15 = lane & 15;
  int w0 = (mTileStart + mt) * 16;
  const float* AT = ATbase + (size_t)e * kT * kT;
  const float* Bb = Bbase + ((size_t)e * kNB + b) * kT * kC;
  float* Db = Dbase + ((size_t)e * kNB + b) * (size_t)outRows * kC;
  const float* ATrow = AT + (size_t)(valid ? (w0 + l15) : l15) * kT;

  // ---- cooperative chunk copy: global (contiguous) -> LDS ----
  auto issue_chunk = [&](int ch, int sel) {
    const float* gsrc = Bb + (size_t)ch * kKCHUNK * kC;  // 16B-aligned, contiguous
#if HAS_ASYNC_LDS
    #pragma unroll
    for (int j = 0; j < kQUADS_PER_THREAD; ++j) {
      int q = tid + j * 128;
      __builtin_amdgcn_global_load_async_to_lds_b128(
          (v4i*)(gsrc + (size_t)q * 4),
          (as3_v4i*)(&sbuf[sel][q * 4]), 0, 0);
    }
#else
    #pragma unroll
    for (int j = 0; j < kQUADS_PER_THREAD; ++j) {
      int q = tid + j * 128;
      float4 v = ((const float4*)gsrc)[q];
      ((float4*)&sbuf[sel][0])[q] = v;
    }
#endif
  };

#if HAS_WMMA_F32K4
  v8f acc0 = {0.f, 0.f, 0.f, 0.f, 0.f, 0.f, 0.f, 0.f};
  v8f acc1 = {0.f, 0.f, 0.f, 0.f, 0.f, 0.f, 0.f, 0.f};
#endif

#if HAS_ASYNC_LDS
  issue_chunk(0, 0);
  issue_chunk(1, 1);
#endif

  for (int ch = 0; ch < kNCHUNK; ++ch) {
    int sel = ch & 1;
    int k0 = ch * kKCHUNK;
#if HAS_ASYNC_LDS
    // async loads complete in order: waiting <= (one chunk's per-wave issue count)
    // guarantees chunk `ch` has fully landed while chunk ch+1 stays in flight.
    if (ch + 1 < kNCHUNK)
      __builtin_amdgcn_s_wait_asynccnt(kQUADS_PER_THREAD);  // constant expr
    else
      __builtin_amdgcn_s_wait_asynccnt(0);
    __syncthreads();
#else
    __syncthreads();
    issue_chunk(ch, sel);
    __syncthreads();
#endif

    if (valid) {
      if (ch + 1 < kNCHUNK)
        __builtin_prefetch(ATrow + k0 + kKCHUNK, 0, 0);  // next A chunk -> caches
      const float* bl0 = &sbuf[sel][0];
      for (int kk = 0; kk < kKCHUNK; kk += 4) {
#if HAS_WMMA_F32K4
        // A fragment 16x4: lanes 0-15 {K=0,K=1}, lanes 16-31 {K=2,K=3} (contiguous float2)
        const float* ap = ATrow + k0 + kk + 2 * half;
        v2f afrag;
        afrag.x = ap[0];
        afrag.y = ap[1];
        // B fragments 4x16 from LDS: lanes 0-15 rows {0,1}, lanes 16-31 rows {2,3}
        const float* bl = bl0 + (kk + 2 * half) * kC;
        v2f bf0, bf1;
        bf0.x = bl[l15];
        bf0.y = bl[kC + l15];
        bf1.x = bl[16 + l15];
        bf1.y = bl[kC + 16 + l15];
        acc0 = __builtin_amdgcn_wmma_f32_16x16x4_f32(false, afrag, false, bf0, (short)0, acc0, false, false);
        acc1 = __builtin_amdgcn_wmma_f32_16x16x4_f32(false, afrag, false, bf1, (short)0, acc1, false, false);
#else
        (void)bl0;
#endif
      }
#if !HAS_WMMA_F32K4
      // scalar fallback (host pass / missing builtin): full dot per element
      for (int idx = lane; idx < 16 * kC; idx += 32) {
        int r = idx >> 5, cc = idx & 31;
        float s = 0.f;
        for (int k = 0; k < kKCHUNK; ++k)
          s += AT[(size_t)(w0 + r) * kT + k0 + k] * bl0[k * kC + cc];
        if (ch == 0) Db[(size_t)(mt * 16 + r) * kC + cc] = s;
        else Db[(size_t)(mt * 16 + r) * kC + cc] += s;
      }
#endif
    }
    __syncthreads();  // everyone done with sbuf[sel] before it is refilled
#if HAS_ASYNC_LDS
    if (ch + 2 < kNCHUNK) issue_chunk(ch + 2, sel);
#endif
  }

#if HAS_WMMA_F32K4
  if (valid) {
    // C/D layout: vgpr i, lanes 0-15 -> (M=i, N=lane); lanes 16-31 -> (M=8+i, N=lane-16)
    int rowBase = mt * 16 + half * 8;
    #pragma unroll
    for (int i = 0; i < 8; ++i) {
      Db[(size_t)(rowBase + i) * kC + l15] = acc0[i];
      Db[(size_t)(rowBase + i) * kC + 16 + l15] = acc1[i];
    }
  }
#endif
}

// ---------------- stage 7: finalize (node GCN scalars + gelu-MLP collapse + gate combine) ----------------
// grid (B, P), block 32 (thread = c)
__global__ void finalize_kernel(const float* __restrict__ ms, const float* __restrict__ ms1,
                                const float* __restrict__ ms2, const float* __restrict__ An,
                                const float* __restrict__ cbuf, const float* __restrict__ gates,
                                float* __restrict__ y) {
  int b = blockIdx.x, p = blockIdx.y, c = threadIdx.x;
  int t = kT - kP + p;
  __shared__ float mrow[kC];
  __shared__ float msn1s[kC];
  float acc = 0.f;
  for (int e = 0; e < kE; ++e) {
    float g = gates[b * kE + e];  // uniform across block
    if (g != 0.f) {
      const float* msb = ms + ((size_t)e * kNB + b) * kT * kC;
      float m0 = msb[(size_t)t * kC + c];
      mrow[c] = m0;
      __syncthreads();
      const float* an = An + (size_t)e * kC * kC;
      float mn1 = 0.f;
      #pragma unroll
      for (int k = 0; k < kC; ++k) mn1 += mrow[k] * an[k * kC + c];
      __syncthreads();
      msn1s[c] = mn1;
      __syncthreads();
      float mn2 = 0.f;
      #pragma unroll
      for (int k = 0; k < kC; ++k) mn2 += msn1s[k] * an[k * kC + c];
      float m1 = ms1[(((size_t)e * kNB + b) * kT + t) * kC + c];
      float m2 = ms2[(((size_t)e * kNB + b) * kP + p) * kC + c];
      const float* cb = cbuf + (size_t)e * kCBUF;
      const float* U0 = cb, *U1 = cb + 32, *U2 = cb + 64, *Vs = cb + 96;
      const float* P0 = cb + 128, *P1 = cb + 160, *P2 = cb + 192;
      const float* Rb = cb + 224 + c * kD;
      const float* owt = cb + 1248, *own = cb + 1280;
      float a = cb[1312];  // out_b
      #pragma unroll 4
      for (int d = 0; d < kD; ++d) {
        float st = m0 * U0[d] + m1 * U1[d] + m2 * U2[d] + Vs[d];
        a += gelu_tanh(st) * owt[d];
        float sn = m0 * P0[d] + mn1 * P1[d] + mn2 * P2[d] + Rb[d];
        a += gelu_tanh(sn) * own[d];
      }
      acc += g * a;
      __syncthreads();
    }
  }
  y[((size_t)b * kP + p) * kC + c] = acc;
}

// ---------------- launcher ----------------
extern "C" void kernel_launch(void* const* d_in, const int* in_sizes, int n_in,
                              void* d_out, int out_size, void* d_ws, size_t ws_size,
                              hipStream_t stream) {
  (void)in_sizes; (void)n_in; (void)out_size; (void)ws_size;
  const float* x       = (const float*)d_in[0];
  const float* noise   = (const float*)d_in[1];
  const float* w_gate  = (const float*)d_in[2];
  const float* w_noise = (const float*)d_in[3];
  const float* tv1     = (const float*)d_in[4];
  const float* tv2     = (const float*)d_in[5];
  const float* nv1     = (const float*)d_in[6];
  const float* nv2     = (const float*)d_in[7];
  const float* start_w = (const float*)d_in[8];
  const float* start_b = (const float*)d_in[9];
  const float* t_mlp_w = (const float*)d_in[10];
  const float* t_mlp_b = (const float*)d_in[11];
  const float* n_mlp_w = (const float*)d_in[12];
  const float* n_mlp_b = (const float*)d_in[13];
  const float* out_w   = (const float*)d_in[14];
  const float* out_b   = (const float*)d_in[15];
  float* y = (float*)d_out;

  float* ws = (float*)d_ws;
  size_t off = 0;
  float* amp   = ws + off; off += (size_t)kNB * kFB;
  float* gates = ws + off; off += (size_t)kNB * kE;
  float* AT    = ws + off; off += (size_t)kE * kT * kT;
  float* An    = ws + off; off += (size_t)kE * kC * kC;
  float* ncs1  = ws + off; off += (size_t)kE * kC;
  float* ncs2  = ws + off; off += (size_t)kE * kC;
  float* cbuf  = ws + off; off += (size_t)kE * kCBUF;
  float* msbuf = ws + off; off += (size_t)kE * kNB * kT * kC;
  float* ms1   = ws + off; off += (size_t)kE * kNB * kT * kC;
  float* ms2   = ws + off; off += (size_t)kE * kNB * kP * kC;

  // routing
  dft_amp_kernel<<<dim3(kNB, kFB), 32, 0, stream>>>(x, amp);
  gating_kernel<<<1, 32, 0, stream>>>(amp, noise, w_gate, w_noise, gates);

  // adjacency + folded constants
  build_At_kernel<<<dim3(kE, kT), 256, 0, stream>>>(tv1, tv2, AT);
  build_An_kernel<<<kE, 32, 0, stream>>>(nv1, nv2, An, ncs1, ncs2);
  precompute_kernel<<<kE, 32, 0, stream>>>(start_w, start_b, t_mlp_w, t_mlp_b,
                                           n_mlp_w, n_mlp_b, out_w, out_b,
                                           ncs1, ncs2, cbuf);

  // multi-scale inputs
  build_ms_kernel<<<dim3((kT * kC) / 256, kNB, kE), 256, 0, stream>>>(x, msbuf);

  // dominant batched GEMMs on WMMA (B staged in LDS via async copies):
  // ms1 = A_t^T @ ms (full T rows), ms2 = A_t^T @ ms1 (last P rows only)
  ms_matmul_wmma<<<dim3(kNB, 9, kE), 128, 0, stream>>>(AT, msbuf, ms1, 0, 33, kT);
  ms_matmul_wmma<<<dim3(kNB, 2, kE), 128, 0, stream>>>(AT, ms1, ms2, 27, 6, kP);

  // node-GCN scalars + gelu MLP collapse + gate-weighted combine
  finalize_kernel<<<dim3(kNB, kP), 32, 0, stream>>>(msbuf, ms1, ms2, An, cbuf, gates, y);
}